// NeuronLfm2ShortConv_11089605558863
// MI455X (gfx1250) — compile-verified
//
#include <hip/hip_runtime.h>

typedef __attribute__((ext_vector_type(2))) float v2f;
typedef __attribute__((ext_vector_type(8))) float v8f;

#define Hdim  2048
#define Sdim  4096
#define Mrows 8192            // 2 * 4096
#define Ksz   2048
#define KS    16              // K-step staged through LDS
#define APAD  20              // padded row stride (floats): 16B-aligned rows, conflict-free

// ---- CDNA5 async copy: 16B per lane, global -> LDS, tracked by ASYNCcnt ----
__device__ __forceinline__ void async_copy16(const float* gptr, const float* lptr) {
  asm volatile("global_load_async_to_lds_b128 %0, %1, off"
               :: "v"((unsigned)(uintptr_t)lptr),
                  "v"((unsigned long long)(uintptr_t)gptr)
               : "memory");
}
#define WAIT_ASYNC(n) asm volatile("s_wait_asynccnt " #n ::: "memory")

// =====================================================================
// Kernel 1: fused in-projection GEMM.
//   BCx[m, g*H+n] = sum_k hs[m,k] * Win[g*H+n, k],  g in {B=0, C=1, x=2}
// Epilogue writes ws_bx = B*x and ws_c = C  (each Mrows x H).
// 256 thr (8 waves as 4Mx2N), block tile 128x64, wave tile 32x32 x3 groups.
// Double-buffered LDS fed by global_load_async_to_lds_b128.
// =====================================================================
#define BM1 128
#define BN1 64

__global__ __launch_bounds__(256)
void k1_inproj(const float* __restrict__ A, const float* __restrict__ Win,
               float* __restrict__ ws_bx, float* __restrict__ ws_c) {
  __shared__ __align__(16) float As[2][BM1][APAD];      // [buf][m][k]
  __shared__ __align__(16) float Bs[2][3][BN1][APAD];   // [buf][g][n][k]

  const int tid   = threadIdx.x;
  const int lane  = tid & 31;
  const int wave  = tid >> 5;     // 0..7
  const int waveM = wave >> 1;    // 0..3
  const int waveN = wave & 1;     // 0..1
  const int l16   = lane & 15;
  const int lhalf = lane >> 4;    // 0 or 1

  const int m0 = blockIdx.y * BM1;
  const int n0 = blockIdx.x * BN1;

  v8f acc[3][2][2];
  const v8f vzero = {};
  #pragma unroll
  for (int g = 0; g < 3; ++g)
    #pragma unroll
    for (int i = 0; i < 2; ++i)
      #pragma unroll
      for (int j = 0; j < 2; ++j)
        acc[g][i][j] = vzero;

  // per-thread 16B-transfer coords (5 async instructions per batch)
  const int aRow = tid >> 2;            // 0..63 (two chunks: aRow, aRow+64)
  const int aCol = (tid & 3) << 2;      // 0,4,8,12
  const int bN   = tid >> 2;            // 0..63
  const int bK   = (tid & 3) << 2;      // 0,4,8,12

  auto issue = [&](int k0, int buf) {
    async_copy16(&A[(size_t)(m0 + aRow)      * Ksz + k0 + aCol], &As[buf][aRow][aCol]);
    async_copy16(&A[(size_t)(m0 + aRow + 64) * Ksz + k0 + aCol], &As[buf][aRow + 64][aCol]);
    #pragma unroll
    for (int g = 0; g < 3; ++g)
      async_copy16(&Win[(size_t)(g * Hdim + n0 + bN) * Ksz + k0 + bK],
                   &Bs[buf][g][bN][bK]);
  };

  issue(0, 0);
  for (int i = 0; i < Ksz / KS; ++i) {
    const int buf = i & 1;
    if (i + 1 < Ksz / KS) {
      issue((i + 1) * KS, buf ^ 1);   // prefetch next tile into other buffer
      WAIT_ASYNC(5);                  // 5 outstanding => batch i fully in LDS
    } else {
      WAIT_ASYNC(0);
    }
    __syncthreads();                  // batch i visible to all waves

    #pragma unroll
    for (int kb = 0; kb < KS / 4; ++kb) {
      const int kc = kb * 4 + 2 * lhalf;     // f32 WMMA lane K base
      v2f afr[2];
      #pragma unroll
      for (int im = 0; im < 2; ++im) {
        const float* ap = &As[buf][waveM * 32 + im * 16 + l16][kc];
        afr[im][0] = ap[0];
        afr[im][1] = ap[1];
      }
      #pragma unroll
      for (int g = 0; g < 3; ++g) {
        #pragma unroll
        for (int jn = 0; jn < 2; ++jn) {
          const float* bp = &Bs[buf][g][waveN * 32 + jn * 16 + l16][kc];
          v2f bfr;
          bfr[0] = bp[0];
          bfr[1] = bp[1];
          #pragma unroll
          for (int im = 0; im < 2; ++im) {
            acc[g][im][jn] = __builtin_amdgcn_wmma_f32_16x16x4_f32(
                false, afr[im], false, bfr, (short)0, acc[g][im][jn],
                false, false);
          }
        }
      }
    }
    __syncthreads();                  // done reading buf before it is refilled
  }

  // Epilogue: Bx = B*x, C passthrough.
  #pragma unroll
  for (int im = 0; im < 2; ++im) {
    #pragma unroll
    for (int jn = 0; jn < 2; ++jn) {
      const int col     = n0 + waveN * 32 + jn * 16 + l16;
      const int rowBase = m0 + waveM * 32 + im * 16 + 8 * lhalf;
      #pragma unroll
      for (int i = 0; i < 8; ++i) {
        const size_t idx = (size_t)(rowBase + i) * Hdim + col;
        ws_bx[idx] = acc[0][im][jn][i] * acc[2][im][jn][i];
        ws_c[idx]  = acc[1][im][jn][i];
      }
    }
  }
}

// =====================================================================
// Kernel 2: causal depthwise conv (L=3) + multiply by C, in place over C.
//   y[m,n] = C[m,n] * (w[n,2]*Bx[s] + w[n,1]*Bx[s-1] + w[n,0]*Bx[s-2])
// =====================================================================
__global__ __launch_bounds__(256)
void k2_conv(const float* __restrict__ bx, float* __restrict__ c_io,
             const float* __restrict__ w) {
  const int idx = blockIdx.x * 256 + threadIdx.x;   // exact grid, no tail
  const int n = idx & (Hdim - 1);
  const int m = idx >> 11;                          // Hdim == 2^11
  const int s = m & (Sdim - 1);
  const float w0 = w[n * 3 + 0];
  const float w1 = w[n * 3 + 1];
  const float w2 = w[n * 3 + 2];
  float v = w2 * bx[idx];
  if (s >= 1) v += w1 * bx[idx - Hdim];
  if (s >= 2) v += w0 * bx[idx - 2 * Hdim];
  c_io[idx] *= v;
}

// =====================================================================
// Kernel 3: out-projection GEMM.  out[m,n] = sum_k y[m,k] * Wout[n,k]
// 256 thr (8 waves as 2Mx4N), block tile 128x128, wave tile 64x32.
// Double-buffered LDS fed by global_load_async_to_lds_b128.
// =====================================================================
#define BM3 128
#define BN3 128

__global__ __launch_bounds__(256)
void k3_outproj(const float* __restrict__ Y, const float* __restrict__ Wout,
                float* __restrict__ Out) {
  __shared__ __align__(16) float As[2][BM3][APAD];   // [buf][m][k]
  __shared__ __align__(16) float Bs[2][BN3][APAD];   // [buf][n][k]

  const int tid   = threadIdx.x;
  const int lane  = tid & 31;
  const int wave  = tid >> 5;
  const int waveM = wave >> 2;    // 0..1 (64 rows each)
  const int waveN = wave & 3;     // 0..3 (32 cols each)
  const int l16   = lane & 15;
  const int lhalf = lane >> 4;

  const int m0 = blockIdx.y * BM3;
  const int n0 = blockIdx.x * BN3;

  v8f acc[4][2];
  const v8f vzero = {};
  #pragma unroll
  for (int i = 0; i < 4; ++i)
    #pragma unroll
    for (int j = 0; j < 2; ++j)
      acc[i][j] = vzero;

  const int aRow = tid >> 2;
  const int aCol = (tid & 3) << 2;
  const int bN = tid >> 2;          // 0..63 (two chunks: bN, bN+64)
  const int bK = (tid & 3) << 2;

  auto issue = [&](int k0, int buf) {
    async_copy16(&Y[(size_t)(m0 + aRow)      * Ksz + k0 + aCol], &As[buf][aRow][aCol]);
    async_copy16(&Y[(size_t)(m0 + aRow + 64) * Ksz + k0 + aCol], &As[buf][aRow + 64][aCol]);
    async_copy16(&Wout[(size_t)(n0 + bN)      * Ksz + k0 + bK], &Bs[buf][bN][bK]);
    async_copy16(&Wout[(size_t)(n0 + bN + 64) * Ksz + k0 + bK], &Bs[buf][bN + 64][bK]);
  };

  issue(0, 0);
  for (int i = 0; i < Ksz / KS; ++i) {
    const int buf = i & 1;
    if (i + 1 < Ksz / KS) {
      issue((i + 1) * KS, buf ^ 1);
      WAIT_ASYNC(4);                  // 4 outstanding => batch i fully in LDS
    } else {
      WAIT_ASYNC(0);
    }
    __syncthreads();

    #pragma unroll
    for (int kb = 0; kb < KS / 4; ++kb) {
      const int kc = kb * 4 + 2 * lhalf;
      v2f afr[4];
      #pragma unroll
      for (int im = 0; im < 4; ++im) {
        const float* ap = &As[buf][waveM * 64 + im * 16 + l16][kc];
        afr[im][0] = ap[0];
        afr[im][1] = ap[1];
      }
      #pragma unroll
      for (int jn = 0; jn < 2; ++jn) {
        const float* bp = &Bs[buf][waveN * 32 + jn * 16 + l16][kc];
        v2f bfr;
        bfr[0] = bp[0];
        bfr[1] = bp[1];
        #pragma unroll
        for (int im = 0; im < 4; ++im) {
          acc[im][jn] = __builtin_amdgcn_wmma_f32_16x16x4_f32(
              false, afr[im], false, bfr, (short)0, acc[im][jn], false, false);
        }
      }
    }
    __syncthreads();
  }

  #pragma unroll
  for (int im = 0; im < 4; ++im) {
    #pragma unroll
    for (int jn = 0; jn < 2; ++jn) {
      const int col     = n0 + waveN * 32 + jn * 16 + l16;
      const int rowBase = m0 + waveM * 64 + im * 16 + 8 * lhalf;
      #pragma unroll
      for (int i = 0; i < 8; ++i) {
        Out[(size_t)(rowBase + i) * Hdim + col] = acc[im][jn][i];
      }
    }
  }
}

// =====================================================================
extern "C" void kernel_launch(void* const* d_in, const int* in_sizes, int n_in,
                              void* d_out, int out_size, void* d_ws, size_t ws_size,
                              hipStream_t stream) {
  const float* hs    = (const float*)d_in[0];  // (2, 4096, 2048)
  const float* Win   = (const float*)d_in[1];  // (6144, 2048)
  const float* convw = (const float*)d_in[2];  // (2048, 1, 3)
  const float* Wout  = (const float*)d_in[3];  // (2048, 2048)
  float* out = (float*)d_out;                  // (2, 4096, 2048)

  float* ws_bx = (float*)d_ws;                       // Mrows*Hdim floats
  float* ws_c  = ws_bx + (size_t)Mrows * Hdim;       // Mrows*Hdim floats

  // K1: fused in-projection (B, C, x) -> Bx, C
  k1_inproj<<<dim3(Hdim / BN1, Mrows / BM1), 256, 0, stream>>>(hs, Win, ws_bx, ws_c);
  // K2: causal conv-3 + C multiply (in place over ws_c)
  k2_conv<<<(Mrows * Hdim) / 256, 256, 0, stream>>>(ws_bx, ws_c, convw);
  // K3: out-projection
  k3_outproj<<<dim3(Hdim / BN3, Mrows / BM3), 256, 0, stream>>>(ws_c, Wout, out);
}